// HeteroGNN_89369679495191
// MI455X (gfx1250) — compile-verified
//
#include <hip/hip_runtime.h>
#include <math.h>

#define N_NODES 100000
#define N_REL 7
#define NEDGE 1600000
#define D_IN 128
#define D_H 256
#define D_C 128   // D_H/2
#define EPS 1e-5f

typedef __attribute__((ext_vector_type(2))) float v2f;
typedef __attribute__((ext_vector_type(8))) float v8f;
typedef __attribute__((ext_vector_type(4))) unsigned int u32x4;
typedef __attribute__((ext_vector_type(8))) int i32x8;
typedef __attribute__((ext_vector_type(4))) int i32x4;

#if __has_builtin(__builtin_amdgcn_tensor_load_to_lds)
#define HAVE_TDM 1
#else
#define HAVE_TDM 0
#endif

// ---------------------------------------------------------------- utilities

__global__ void zero_kernel(float* __restrict__ p, long n) {
  long i = (long)blockIdx.x * blockDim.x + threadIdx.x;
  long stride = (long)gridDim.x * blockDim.x;
  for (; i < n; i += stride) p[i] = 0.f;
}

// Wsum[k,n] = sum_r W[r,k,n]; bsum[n] = sum_r b[r,n]
__global__ void sum_weights_kernel(const float* __restrict__ W,
                                   const float* __restrict__ b,
                                   int K, int N,
                                   float* __restrict__ Wsum,
                                   float* __restrict__ bsum) {
  int i = blockIdx.x * blockDim.x + threadIdx.x;
  int total = K * N;
  if (i < total) {
    float s = 0.f;
    for (int r = 0; r < N_REL; ++r) s += W[(long)r * total + i];
    Wsum[i] = s;
  }
  if (i < N) {
    float s = 0.f;
    for (int r = 0; r < N_REL; ++r) s += b[r * N + i];
    bsum[i] = s;
  }
}

// ------------------------------------------------------- edge scatter (mean)
// one wave per 4 edges: agg[dst,:] += x[src,:]; deg[dst] += 1
// source rows are prefetched (global_prefetch_b8) before the atomic phase.
__global__ void scatter_kernel(const float* __restrict__ x,
                               const int* __restrict__ src,
                               const int* __restrict__ dst,
                               float* __restrict__ agg,
                               float* __restrict__ deg,
                               int nEdges, int D) {
  int w    = (int)((blockIdx.x * blockDim.x + threadIdx.x) >> 5);
  int lane = threadIdx.x & 31;
  int base = w * 4;
  if (base >= nEdges) return;
  int4 s4 = *(const int4*)(src + base);
  int4 d4 = *(const int4*)(dst + base);
  int ss[4] = {s4.x, s4.y, s4.z, s4.w};
  int dd[4] = {d4.x, d4.y, d4.z, d4.w};

  // prefetch the four gather rows (one lane per 128B line)
  int lines = D >> 5;  // (D*4)/128
  if (lane < lines) {
#pragma unroll
    for (int i = 0; i < 4; ++i)
      __builtin_prefetch(x + (long)ss[i] * D + (lane << 5), 0, 1);
  }

#pragma unroll
  for (int i = 0; i < 4; ++i) {
    const float4* xr = (const float4*)(x + (long)ss[i] * D);
    float* ar = agg + (long)dd[i] * D;
    for (int c = lane * 4; c < D; c += 128) {
      float4 v = xr[c >> 2];
      atomicAdd(ar + c + 0, v.x);
      atomicAdd(ar + c + 1, v.y);
      atomicAdd(ar + c + 2, v.z);
      atomicAdd(ar + c + 3, v.w);
    }
    if (lane == 0) atomicAdd(deg + dd[i], 1.0f);
  }
}

// -------------------------------------------------------- fp32 WMMA GEMM
// C[M,N] (+)= rowscale(A[M,K]) @ B[K,N]   (B row-major: B[k*N+n])
// grid = (M/16, N/64), blockDim = 128 (4 waves; wave w -> ntile = by*4+w).
// The 16xK A tile (contiguous rows) is staged into LDS once per block via the
// Tensor Data Mover (1-D descriptor), then shared by all 4 waves.
// accumulate=0: C initialized to bias (or 0); accumulate=1: C read-modify-write
// postG != null: epilogue  c = relu(c) * postG[n]*rsqrt(1+eps) + postB[n]
__global__ void gemm_wmma_kernel(const float* __restrict__ A, int K,
                                 const float* __restrict__ B, int N,
                                 const float* __restrict__ rowScaleDeg,
                                 const float* __restrict__ bias,
                                 int accumulate,
                                 const float* __restrict__ postG,
                                 const float* __restrict__ postB,
                                 float* __restrict__ C) {
  __shared__ __align__(16) float ldsA[16 * 256];  // 16KB, K <= 256

  int lane  = threadIdx.x & 31;
  int wave  = threadIdx.x >> 5;
  int mtile = blockIdx.x;
  int ntile = blockIdx.y * 4 + wave;
  int row   = mtile * 16 + (lane & 15);   // A row for this lane's fragment
  int col   = ntile * 16 + (lane & 15);   // B/C column for this lane
  int khalf = (lane >> 4) << 1;           // lanes 0-15 -> K+0..1, 16-31 -> K+2..3
  int mhalf = (lane >> 4) << 3;           // C rows: VGPR j -> M = j (+8 for hi lanes)

  const float* Atile = A + (long)mtile * 16 * K;  // 16 contiguous rows
  unsigned int nelem = 16u * (unsigned int)K;

#if HAVE_TDM
  if (threadIdx.x < 32) {  // wave 0 issues the TDM transfer
    unsigned long long ga = (unsigned long long)(const void*)Atile;
    unsigned int la = (unsigned int)(unsigned long long)(void*)ldsA;
    u32x4 g0;
    g0.x = 1u;                                                 // count=1
    g0.y = la;                                                 // lds_addr (bytes)
    g0.z = (unsigned int)ga;                                   // global_addr[31:0]
    g0.w = (unsigned int)((ga >> 32) & 0x01FFFFFFu) | (2u << 30);  // [56:32] + type=2
    i32x8 g1;
    g1[0] = 2 << 16;                                  // data_size = 4 bytes
    g1[1] = (int)(nelem << 16);                       // tensor_dim0[15:0]  @63:48
    g1[2] = (int)((nelem >> 16) & 0xFFFFu) | (1 << 16); // tensor_dim0[31:16], tensor_dim1=1
    g1[3] = (int)(nelem << 16);                       // tile_dim0          @127:112
    g1[4] = 0;                                        // tile_dim1/2 unused
    g1[5] = (int)nelem;                               // tensor_dim0_stride lo32
    g1[6] = 0;
    g1[7] = 0;
    i32x4 gz = (i32x4){0, 0, 0, 0};
    i32x8 gz8 = (i32x8){0, 0, 0, 0, 0, 0, 0, 0};
    __builtin_amdgcn_tensor_load_to_lds(g0, g1, gz, gz, gz8, 0);
    __builtin_amdgcn_s_wait_tensorcnt(0);
  }
  __syncthreads();
#else
  for (int idx = threadIdx.x * 4; idx < 16 * K; idx += 128 * 4)
    *(float4*)&ldsA[idx] = *(const float4*)&Atile[idx];
  __syncthreads();
#endif

  float rs = 1.0f;
  if (rowScaleDeg) rs = 1.0f / fmaxf(rowScaleDeg[row], 1.0f);

  v8f c;
  if (accumulate) {
#pragma unroll
    for (int j = 0; j < 8; ++j)
      c[j] = C[(long)(mtile * 16 + j + mhalf) * N + col];
  } else if (bias) {
    float bv = bias[col];
#pragma unroll
    for (int j = 0; j < 8; ++j) c[j] = bv;
  } else {
#pragma unroll
    for (int j = 0; j < 8; ++j) c[j] = 0.f;
  }

  const float* Alds = ldsA + (lane & 15) * K;
  for (int k = 0; k < K; k += 4) {
    int kk = k + khalf;
    v2f a, b;
    a.x = rs * Alds[kk];
    a.y = rs * Alds[kk + 1];
    b.x = B[(long)kk * N + col];
    b.y = B[(long)(kk + 1) * N + col];
    c = __builtin_amdgcn_wmma_f32_16x16x4_f32(false, a, false, b,
                                              (short)0, c, false, false);
  }

  if (postG) {
    float g  = postG[col] * rsqrtf(1.0f + EPS);
    float pb = postB[col];
#pragma unroll
    for (int j = 0; j < 8; ++j) c[j] = fmaxf(c[j], 0.f) * g + pb;
  }

#pragma unroll
  for (int j = 0; j < 8; ++j)
    C[(long)(mtile * 16 + j + mhalf) * N + col] = c[j];
}

// ------------------------------------------------- LayerNorm(x/7) + ReLU
// one wave per row of D_H=256 (8 elems/lane), wave32 shfl reductions
__global__ void ln_relu_kernel(const float* __restrict__ acc,
                               const float* __restrict__ g,
                               const float* __restrict__ b,
                               float* __restrict__ out) {
  int wave = (int)((blockIdx.x * blockDim.x + threadIdx.x) >> 5);
  int lane = threadIdx.x & 31;
  if (wave >= N_NODES) return;
  const float* xr = acc + (long)wave * D_H;
  float v[8];
  float sum = 0.f;
#pragma unroll
  for (int j = 0; j < 8; ++j) {
    v[j] = xr[lane + 32 * j] * (1.0f / N_REL);
    sum += v[j];
  }
  for (int off = 16; off > 0; off >>= 1) sum += __shfl_xor(sum, off, 32);
  float mu = sum * (1.0f / D_H);
  float var = 0.f;
#pragma unroll
  for (int j = 0; j < 8; ++j) { float d = v[j] - mu; var += d * d; }
  for (int off = 16; off > 0; off >>= 1) var += __shfl_xor(var, off, 32);
  float inv = rsqrtf(var * (1.0f / D_H) + EPS);
  float* orow = out + (long)wave * D_H;
#pragma unroll
  for (int j = 0; j < 8; ++j) {
    int ci = lane + 32 * j;
    float y = (v[j] - mu) * inv * g[ci] + b[ci];
    orow[ci] = fmaxf(y, 0.f);
  }
}

// ------------------------------------------- final skinny linear 128 -> 2
__global__ void final_linear_kernel(const float* __restrict__ z,
                                    const float* __restrict__ W,   // (128,2)
                                    const float* __restrict__ bias,
                                    float* __restrict__ out) {
  int wave = (int)((blockIdx.x * blockDim.x + threadIdx.x) >> 5);
  int lane = threadIdx.x & 31;
  if (wave >= N_NODES) return;
  const float* zr = z + (long)wave * D_C;
  float s0 = 0.f, s1 = 0.f;
#pragma unroll
  for (int j = 0; j < 4; ++j) {
    int k = lane + 32 * j;
    float v = zr[k];
    s0 += v * W[k * 2 + 0];
    s1 += v * W[k * 2 + 1];
  }
  for (int off = 16; off > 0; off >>= 1) {
    s0 += __shfl_xor(s0, off, 32);
    s1 += __shfl_xor(s1, off, 32);
  }
  if (lane == 0) {
    out[(long)wave * 2 + 0] = s0 + bias[0];
    out[(long)wave * 2 + 1] = s1 + bias[1];
  }
}

// --------------------------------------------------------------- launcher

extern "C" void kernel_launch(void* const* d_in, const int* in_sizes, int n_in,
                              void* d_out, int out_size, void* d_ws, size_t ws_size,
                              hipStream_t stream) {
  (void)in_sizes; (void)n_in; (void)out_size; (void)ws_size;

  const float* features = (const float*)d_in[0];
  const int*   edges    = (const int*)d_in[1];    // (7, 2, E)
  const float* Wself1   = (const float*)d_in[2];  // (7,128,256)
  const float* Wneigh1  = (const float*)d_in[3];
  const float* b1       = (const float*)d_in[4];
  const float* Wself2   = (const float*)d_in[5];  // (7,256,256)
  const float* Wneigh2  = (const float*)d_in[6];
  const float* b2       = (const float*)d_in[7];
  const float* ln_g     = (const float*)d_in[8];  // (2,256)
  const float* ln_b     = (const float*)d_in[9];
  const float* Wc1      = (const float*)d_in[10]; // (256,128)
  const float* bc1      = (const float*)d_in[11];
  const float* bn_g     = (const float*)d_in[12];
  const float* bn_b     = (const float*)d_in[13];
  const float* Wc2      = (const float*)d_in[14]; // (128,2)
  const float* bc2      = (const float*)d_in[15];
  float* out = (float*)d_out;

  // workspace carve-up
  char* ws = (char*)d_ws;
  float* Wsum = (float*)ws;  ws += (size_t)D_H * D_H * sizeof(float);   // 256KB
  float* bsum = (float*)ws;  ws += 1024;
  float* agg  = (float*)ws;  ws += (size_t)N_NODES * D_H * sizeof(float);
  float* deg  = (float*)ws;  ws += (size_t)N_NODES * sizeof(float);
  float* acc  = (float*)ws;  ws += (size_t)N_NODES * D_H * sizeof(float);
  float* h    = (float*)ws;  ws += (size_t)N_NODES * D_H * sizeof(float);
  float* z    = agg;  // agg is dead once the classifier starts

  auto run_layer = [&](const float* x, int K, const float* Wself,
                       const float* Wneigh, const float* bias, int lnIdx,
                       float* outbuf) {
    int tot = K * D_H;
    sum_weights_kernel<<<(tot + 255) / 256, 256, 0, stream>>>(Wself, bias, K, D_H, Wsum, bsum);
    dim3 g(N_NODES / 16, D_H / 64);
    // self term (+ summed bias), overwrite-init
    gemm_wmma_kernel<<<g, 128, 0, stream>>>(x, K, Wsum, D_H, nullptr, bsum, 0,
                                            nullptr, nullptr, acc);
    for (int r = 0; r < N_REL; ++r) {
      zero_kernel<<<2048, 256, 0, stream>>>(agg, (long)N_NODES * K);
      zero_kernel<<<256, 256, 0, stream>>>(deg, (long)N_NODES);
      const int* src = edges + (long)r * 2 * NEDGE;
      const int* dst = src + NEDGE;
      scatter_kernel<<<(NEDGE / 4 + 7) / 8, 256, 0, stream>>>(x, src, dst, agg, deg, NEDGE, K);
      // mean_r @ Wneigh_r, 1/deg fused into A-fragment load, accumulate into acc
      gemm_wmma_kernel<<<g, 128, 0, stream>>>(agg, K, Wneigh + (long)r * K * D_H, D_H,
                                              deg, nullptr, 1, nullptr, nullptr, acc);
    }
    ln_relu_kernel<<<(N_NODES + 7) / 8, 256, 0, stream>>>(acc, ln_g + lnIdx * D_H,
                                                          ln_b + lnIdx * D_H, outbuf);
  };

  run_layer(features, D_IN, Wself1, Wneigh1, b1, 0, h);
  run_layer(h,        D_H,  Wself2, Wneigh2, b2, 1, h);

  // classifier: z = BN(relu(h @ Wc1 + bc1));  out = z @ Wc2 + bc2
  dim3 gc(N_NODES / 16, D_C / 64);
  gemm_wmma_kernel<<<gc, 128, 0, stream>>>(h, D_H, Wc1, D_C, nullptr, bc1, 0,
                                           bn_g, bn_b, z);
  final_linear_kernel<<<(N_NODES + 7) / 8, 256, 0, stream>>>(z, Wc2, bc2, out);
}